// CASTransformer_46943992545732
// MI455X (gfx1250) — compile-verified
//
#include <hip/hip_runtime.h>
#include <math.h>
#include <stdint.h>

typedef __attribute__((ext_vector_type(2))) float v2f;
typedef __attribute__((ext_vector_type(8))) float v8f;

// ---------------------------------------------------------------------------
// Async staging of one 128 x 64 W chunk into LDS via GLOBAL_LOAD_ASYNC_TO_LDS.
// Each of the 256 threads issues 8 b128 copies (2048 x 16B = 32 KB chunk).
// Row stride is 68 floats (272 B, 16B-aligned) so every b128 LDS write is
// naturally aligned; tracked by ASYNCcnt.
// ---------------------------------------------------------------------------
__device__ __forceinline__ void stage_w_async(const float* __restrict__ W, int K,
                                              int blockN, int kc,
                                              float (*sWbuf)[68], int tid)
{
    #pragma unroll
    for (int t = 0; t < 8; ++t) {
        const int c = tid + t * 256;          // 0..2047 vec4 chunks
        const int n = c >> 4;                 // 16 vec4 per 64-float row
        const int k = (c & 15) << 2;
        const float* src = W + (size_t)(blockN + n) * K + kc + k;
        const uint32_t dst = (uint32_t)(uintptr_t)&sWbuf[n][k];  // LDS byte offset
        asm volatile("global_load_async_to_lds_b128 %0, %1, off"
                     :: "v"(dst), "v"(src)
                     : "memory");
    }
}

__device__ __forceinline__ void wait_async_all()
{
    asm volatile("s_wait_asynccnt 0x0" ::: "memory");
}

// ---------------------------------------------------------------------------
// GEMM: Y[m, n] = sum_k X[m, xcol0+k] * W[n*K + k] + bias[n] (+ residual[m,n])
//   W is row-major (N x K), i.e. torch Linear weight, computing X @ W^T + b.
//   Block = 256 threads = 8 wave32s; block tile = 128 (M) x 128 (N).
//   Wave w computes rows [blockM + 16w, +16) x 128 cols = 8 WMMA 16x16 tiles.
//   Uses V_WMMA_F32_16X16X4_F32 (exact fp32 matrix path on CDNA5) with
//   double-buffered async W staging: chunk i+1 streams into LDS while the
//   128-WMMA burst for chunk i executes.
//   Requires: M % 128 == 0 (grid.x), N % 128 == 0 (grid.y), K % 64 == 0.
// ---------------------------------------------------------------------------
__global__ __launch_bounds__(256) void wmma_gemm_bias_kernel(
    const float* __restrict__ X, int ldx, int xcol0,
    const float* __restrict__ W,
    const float* __restrict__ bias,
    const float* __restrict__ residual,   // may be nullptr; shape/ld same as Y
    float* __restrict__ Y, int ldy,
    int K)
{
    constexpr int KC = 64;
    __shared__ float sW[2][128][KC + 4];  // [buf][n][k], 16B-aligned rows

    const int tid    = threadIdx.x;
    const int lane   = tid & 31;
    const int wv     = tid >> 5;
    const int blockM = blockIdx.x * 128;
    const int blockN = blockIdx.y * 128;

    // A-matrix 16x4 f32 layout: lanes 0-15 hold M=lane, K=k0+{0,1};
    // lanes 16-31 hold M=lane-16, K=k0+{2,3}.
    const int mlane = lane & 15;
    const int koff  = (lane >> 4) * 2;    // 0 or 2
    const int rowA  = blockM + wv * 16 + mlane;

    v8f acc[8] = {};

    // Prime the pipeline: stage chunk 0 into buffer 0.
    stage_w_async(W, K, blockN, 0, sW[0], tid);

    int it = 0;
    for (int kc = 0; kc < K; kc += KC, ++it) {
        wait_async_all();                 // this wave's staging done
        __syncthreads();                  // chunk visible from all waves
        if (kc + KC < K)                  // overlap next chunk with compute
            stage_w_async(W, K, blockN, kc + KC, sW[(it + 1) & 1], tid);

        float (*sWc)[KC + 4] = sW[it & 1];
        const float* xrow = X + (size_t)rowA * ldx + xcol0 + kc + koff;
        // Speculative prefetch of next A chunk (global_prefetch_b8).
        __builtin_prefetch(xrow + KC, 0, 1);

        #pragma unroll
        for (int k0 = 0; k0 < KC; k0 += 4) {
            v2f a;
            a.x = xrow[k0];
            a.y = xrow[k0 + 1];
            #pragma unroll
            for (int t = 0; t < 8; ++t) {
                // B-matrix 4x16 layout mirrors A: lanes 0-15 -> K=k0+{0,1},
                // lanes 16-31 -> K=k0+{2,3}; N = tile col = lane&15.
                const float* wp = &sWc[t * 16 + mlane][k0 + koff];
                v2f b;
                b.x = wp[0];
                b.y = wp[1];
                acc[t] = __builtin_amdgcn_wmma_f32_16x16x4_f32(
                    false, a, false, b, (short)0, acc[t], false, false);
            }
        }
    }

    // C layout: VGPR r, lanes 0-15 -> M = r, lanes 16-31 -> M = r + 8; N = lane&15.
    const int mbase = blockM + wv * 16 + ((lane >> 4) << 3);
    #pragma unroll
    for (int t = 0; t < 8; ++t) {
        const int gn = blockN + t * 16 + mlane;
        const float bv = bias[gn];
        #pragma unroll
        for (int r = 0; r < 8; ++r) {
            const int row = mbase + r;
            const size_t off = (size_t)row * ldy + gn;
            float v = acc[t][r] + bv;
            if (residual) v += residual[off];
            Y[off] = v;
        }
    }
}

// ---------------------------------------------------------------------------
// Dilated local attention, window KERNEL=5, dk=32, heads-per-dilation=4.
// One wave per (b, h, s); lane = channel d in [0,32).
// ---------------------------------------------------------------------------
__global__ __launch_bounds__(256) void local_attn_kernel(
    const float* __restrict__ Qp,       // (B*S, 128)
    const float* __restrict__ Kp,       // (B*S, 128)
    const float* __restrict__ Vp,       // (B*S, 128)
    float* __restrict__ attn_out,       // (B*S, 256), write slice [out_col0, +128)
    int out_col0,
    float* __restrict__ probs_out,      // (B, 8, S, 5) or nullptr
    int head0,
    int dilation, int B, int S)
{
    const int lane = threadIdx.x & 31;
    const int wv   = threadIdx.x >> 5;
    const long wid = (long)blockIdx.x * 8 + wv;
    const long total = (long)B * 4 * S;
    if (wid >= total) return;

    const int b   = (int)(wid / (4L * S));
    const int rem = (int)(wid % (4L * S));
    const int h   = rem / S;
    const int s   = rem % S;
    const int d   = lane;
    const float scale = 0.17677669529663687f;   // 32^-0.5

    const size_t rowQ = (size_t)(b * S + s) * 128 + h * 32 + d;
    const float q = Qp[rowQ];

    float sc[5];
    #pragma unroll
    for (int k = 0; k < 5; ++k) {
        const int pos = s + (k - 2) * dilation;
        float kv = 0.f;
        if (pos >= 0 && pos < S)
            kv = Kp[(size_t)(b * S + pos) * 128 + h * 32 + d];
        float x = q * kv;
        #pragma unroll
        for (int off = 16; off; off >>= 1) x += __shfl_xor(x, off, 32);
        sc[k] = x * scale;
    }

    float mx = sc[0];
    #pragma unroll
    for (int k = 1; k < 5; ++k) mx = fmaxf(mx, sc[k]);
    float e[5];
    float sum = 0.f;
    #pragma unroll
    for (int k = 0; k < 5; ++k) { e[k] = __expf(sc[k] - mx); sum += e[k]; }
    const float inv = 1.f / sum;

    float o = 0.f;
    #pragma unroll
    for (int k = 0; k < 5; ++k) {
        const int pos = s + (k - 2) * dilation;
        float vv = 0.f;
        if (pos >= 0 && pos < S)
            vv = Vp[(size_t)(b * S + pos) * 128 + h * 32 + d];
        o += e[k] * inv * vv;
    }
    attn_out[(size_t)(b * S + s) * 256 + out_col0 + h * 32 + d] = o;

    if (probs_out && lane == 0) {
        const size_t pbase = (((size_t)b * 8 + head0 + h) * S + s) * 5;
        #pragma unroll
        for (int k = 0; k < 5; ++k) probs_out[pbase + k] = e[k] * inv;
    }
}

// ---------------------------------------------------------------------------
// Row LayerNorm over D=256; one block (256 threads) per row.
// ---------------------------------------------------------------------------
__global__ __launch_bounds__(256) void layernorm_kernel(
    const float* __restrict__ Xin,      // (rows, 256), already residual-added
    const float* __restrict__ g,
    const float* __restrict__ bta,
    float* __restrict__ Yout)
{
    __shared__ float s1[256];
    __shared__ float s2[256];
    const int tid = threadIdx.x;
    const size_t row = blockIdx.x;
    const float v = Xin[row * 256 + tid];
    s1[tid] = v;
    s2[tid] = v * v;
    __syncthreads();
    #pragma unroll
    for (int st = 128; st; st >>= 1) {
        if (tid < st) { s1[tid] += s1[tid + st]; s2[tid] += s2[tid + st]; }
        __syncthreads();
    }
    const float mu  = s1[0] * (1.f / 256.f);
    const float var = s2[0] * (1.f / 256.f) - mu * mu;
    const float rs  = rsqrtf(var + 1e-5f);
    Yout[row * 256 + tid] = (v - mu) * rs * g[tid] + bta[tid];
}

// ---------------------------------------------------------------------------
// One dilated-MHA block + residual + LayerNorm.
// ---------------------------------------------------------------------------
static void mha_layer(const float* qk_src, const float* v_src,
                      const float* Wl, const float* bl,      // (2,128,128), (2,128)
                      const float* Wo, const float* bo,      // (256,256), (256)
                      const float* lg, const float* lb,      // (256), (256)
                      const float* resid, float* dst,
                      float* Qp, float* Vp, float* attnbuf, float* ybuf,
                      float* probs, hipStream_t stream)
{
    const int B = 32, S = 2048;
    const size_t MS = (size_t)B * S;                 // 65536 rows
    dim3 blk(256);
    dim3 projGrid((unsigned)(MS / 128), 1);
    dim3 outGrid((unsigned)(MS / 128), 2);
    dim3 lnGrid((unsigned)MS, 1);
    const int attnGrid = (int)((B * 4 * S) / 8);     // 8 waves per block

    for (int i = 0; i < 2; ++i) {
        const int dil = (i == 0) ? 3 : 1;            // DILATIONS = (3, 1)
        const float* Wi = Wl + (size_t)i * 128 * 128;
        const float* bi = bl + (size_t)i * 128;
        // Q (== K) projection of the i-th 128-wide slice.
        wmma_gemm_bias_kernel<<<projGrid, blk, 0, stream>>>(
            qk_src, 256, i * 128, Wi, bi, nullptr, Qp, 128, 128);
        const float* Vbuf = Qp;
        if (v_src != qk_src) {
            wmma_gemm_bias_kernel<<<projGrid, blk, 0, stream>>>(
                v_src, 256, i * 128, Wi, bi, nullptr, Vp, 128, 128);
            Vbuf = Vp;
        }
        local_attn_kernel<<<attnGrid, blk, 0, stream>>>(
            Qp, Qp, Vbuf, attnbuf, i * 128, probs, i * 4, dil, B, S);
    }
    // Output projection + bias + residual, then LayerNorm.
    wmma_gemm_bias_kernel<<<outGrid, blk, 0, stream>>>(
        attnbuf, 256, 0, Wo, bo, resid, ybuf, 256, 256);
    layernorm_kernel<<<lnGrid, blk, 0, stream>>>(ybuf, lg, lb, dst);
}

extern "C" void kernel_launch(void* const* d_in, const int* in_sizes, int n_in,
                              void* d_out, int out_size, void* d_ws, size_t ws_size,
                              hipStream_t stream) {
    (void)in_sizes; (void)n_in; (void)out_size; (void)ws_size;
    const float* q_emb = (const float*)d_in[0];   // (32, 2048, 256)
    const float* s_emb = (const float*)d_in[1];   // (32, 2048, 256)
    /* d_in[2] = lens: unused in eval/shortcut path */
    const float* W_lin = (const float*)d_in[3];   // (3, 2, 128, 128)
    const float* b_lin = (const float*)d_in[4];   // (3, 2, 128)
    const float* W_out = (const float*)d_in[5];   // (3, 256, 256)
    const float* b_out = (const float*)d_in[6];   // (3, 256)
    const float* ln_g  = (const float*)d_in[7];   // (3, 256)
    const float* ln_b  = (const float*)d_in[8];   // (3, 256)

    const int B = 32, S = 2048, D = 256;
    const size_t MS = (size_t)B * S;

    float* out     = (float*)d_out;
    float* z       = out;                          // (B, S, 256)
    float* qscores = out + MS * D;                 // (B, 8, S, 5)

    // Workspace carve-up (needs ~336 MB (320 MiB) of f32 scratch).
    float* ws      = (float*)d_ws;
    float* Qp      = ws;                           // MS * 128
    float* Vp      = Qp + MS * 128;                // MS * 128
    float* attnbuf = Vp + MS * 128;                // MS * 256
    float* ybuf    = attnbuf + MS * 256;           // MS * 256
    float* hq      = ybuf + MS * 256;              // MS * 256
    float* hs      = hq + MS * 256;                // MS * 256

    const size_t WLs = (size_t)2 * 128 * 128;      // per-layer W_lin stride
    const size_t BLs = (size_t)2 * 128;
    const size_t WOs = (size_t)256 * 256;
    const size_t Vs  = 256;

    // Layer 1: self-attention on q_emb -> hq
    mha_layer(q_emb, q_emb, W_lin + 0 * WLs, b_lin + 0 * BLs,
              W_out + 0 * WOs, b_out + 0 * Vs, ln_g + 0 * Vs, ln_b + 0 * Vs,
              q_emb, hq, Qp, Vp, attnbuf, ybuf, nullptr, stream);
    // Layer 2: self-attention on s_emb -> hs (emits q_scores)
    mha_layer(s_emb, s_emb, W_lin + 1 * WLs, b_lin + 1 * BLs,
              W_out + 1 * WOs, b_out + 1 * Vs, ln_g + 1 * Vs, ln_b + 1 * Vs,
              s_emb, hs, Qp, Vp, attnbuf, ybuf, qscores, stream);
    // Layer 3: cross (q=k=hq, v=hs) -> z
    mha_layer(hq, hs, W_lin + 2 * WLs, b_lin + 2 * BLs,
              W_out + 2 * WOs, b_out + 2 * Vs, ln_g + 2 * Vs, ln_b + 2 * Vs,
              hq, z, Qp, Vp, attnbuf, ybuf, nullptr, stream);
}